// POELoss_64939905516221
// MI455X (gfx1250) — compile-verified
//
#include <hip/hip_runtime.h>

// Problem constants (match reference)
constexpr int kS = 5;
constexpr int kB = 256;
constexpr int kD = 512;
constexpr float kMargin = 5.0f;
constexpr float kAlpha  = 1e-4f;
constexpr float kBeta   = 1e-4f;

typedef float v2f __attribute__((ext_vector_type(2)));
typedef float v8f __attribute__((ext_vector_type(8)));

// Workspace byte offsets
constexpr size_t OFF_A1   = 0;                  // [kB*kD] f32 : sigma + mu^2
constexpr size_t OFF_B1   = 512u * 1024u;       // [kB*kD] f32 : inv_sigma
constexpr size_t OFF_B2   = 1024u * 1024u;      // [kB*kD] f32 : -2*mu*inv_sigma
constexpr size_t OFF_C    = 1536u * 1024u;      // [kB*kB] f32 : GEMM result
constexpr size_t OFF_DIST = 1792u * 1024u;      // [kB*kB] f32 : symmetric J-distance
constexpr size_t OFF_Q    = 2048u * 1024u;      // [kB]    f32 : qdiag
constexpr size_t OFF_SC   = 2048u * 1024u + 4096u; // scalars (8B aligned):
//   +0  double trip_sum, +8 float vib_sum, +12 uint trip_count

__global__ void k_init(char* ws) {
    *(double*)(ws + OFF_SC)            = 0.0;
    *(float*)(ws + OFF_SC + 8)         = 0.0f;
    *(unsigned int*)(ws + OFF_SC + 12) = 0u;
}

// One block per row j of mu/log_var: build GEMM operand matrices + row sums.
__global__ __launch_bounds__(256) void k_prep(const float* __restrict__ mu,
                                              const float* __restrict__ lv,
                                              char* ws) {
    const int row = blockIdx.x;
    const int t   = threadIdx.x;
    float* A1 = (float*)(ws + OFF_A1);
    float* B1 = (float*)(ws + OFF_B1);
    float* B2 = (float*)(ws + OFF_B2);
    float* q  = (float*)(ws + OFF_Q);
    float* vs = (float*)(ws + OFF_SC + 8);

    float qp = 0.0f, vp = 0.0f;
    for (int d = t; d < kD; d += 256) {
        const int idx = row * kD + d;
        const float m  = mu[idx];
        const float l  = lv[idx];
        const float s  = expf(l);     // sigma
        const float iv = expf(-l);    // 1/sigma (exact reciprocal via exp)
        A1[idx] = s + m * m;
        B1[idx] = iv;
        B2[idx] = -2.0f * m * iv;
        qp += m * m * iv;
        vp += m * m + s - l - 1.0f;
    }
    __shared__ float sq[256], sv[256];
    sq[t] = qp; sv[t] = vp;
    __syncthreads();
    for (int o = 128; o > 0; o >>= 1) {
        if (t < o) { sq[t] += sq[t + o]; sv[t] += sv[t + o]; }
        __syncthreads();
    }
    if (t == 0) {
        q[row] = sq[0];
        atomicAdd(vs, sv[0]);
    }
}

// WMMA GEMM: C[i,j] = A1·B1^T + mu·B2^T over K=512.
// 8 waves per block, one 16x16 f32 tile per wave, 32 blocks = 256 tiles.
__global__ __launch_bounds__(256) void k_gemm(const float* __restrict__ mu, char* ws) {
    const float* A1 = (const float*)(ws + OFF_A1);
    const float* B1 = (const float*)(ws + OFF_B1);
    const float* B2 = (const float*)(ws + OFF_B2);
    float* C        = (float*)(ws + OFF_C);

    const int wave = threadIdx.x >> 5;
    const int lane = threadIdx.x & 31;
    const int tile = blockIdx.x * 8 + wave;
    const int ti   = tile >> 4;     // row-tile
    const int tj   = tile & 15;     // col-tile
    const int half = lane >> 4;     // K sub-pair: lanes 16-31 carry K+2,K+3
    const int lr   = lane & 15;     // M (for A) / N (for B)

    const float* a1p = A1 + (size_t)(ti * 16 + lr) * kD + 2 * half;
    const float* b1p = B1 + (size_t)(tj * 16 + lr) * kD + 2 * half;
    const float* a2p = mu + (size_t)(ti * 16 + lr) * kD + 2 * half;
    const float* b2p = B2 + (size_t)(tj * 16 + lr) * kD + 2 * half;

    v8f acc = {0.f, 0.f, 0.f, 0.f, 0.f, 0.f, 0.f, 0.f};
    #pragma unroll 4
    for (int k = 0; k < kD; k += 4) {
        v2f a1 = *(const v2f*)(a1p + k);
        v2f b1 = *(const v2f*)(b1p + k);
        v2f a2 = *(const v2f*)(a2p + k);
        v2f b2 = *(const v2f*)(b2p + k);
        // 8 args: (neg_a, A, neg_b, B, c_mod, C, reuse_a, reuse_b)
        acc = __builtin_amdgcn_wmma_f32_16x16x4_f32(false, a1, false, b1,
                                                    (short)0, acc, false, false);
        acc = __builtin_amdgcn_wmma_f32_16x16x4_f32(false, a2, false, b2,
                                                    (short)0, acc, false, false);
    }
    // C/D layout: VGPR r -> row r + 8*half, col = lane&15
    const int col  = tj * 16 + lr;
    const int row0 = ti * 16 + 8 * half;
    #pragma unroll
    for (int r = 0; r < 8; ++r)
        C[(size_t)(row0 + r) * kB + col] = acc[r];
}

// dist[i,j] = 0.5*(C[i,j] + C[j,i] + q[i] + q[j]) - D   (sum_log cancels)
__global__ __launch_bounds__(256) void k_dist(char* ws) {
    const float* C = (const float*)(ws + OFF_C);
    const float* q = (const float*)(ws + OFF_Q);
    float* dist    = (float*)(ws + OFF_DIST);
    const int i = blockIdx.x, j = threadIdx.x;
    dist[i * kB + j] =
        0.5f * (C[i * kB + j] + C[j * kB + i] + q[i] + q[j]) - (float)kD;
}

// Triplet loss: block = i, thread = j, loop over k with rows staged in LDS.
__global__ __launch_bounds__(256) void k_trip(const float* __restrict__ targets,
                                              char* ws) {
    __shared__ float sd[kB], st[kB];
    __shared__ double rs[kB];
    __shared__ unsigned int rc[kB];
    const int i = blockIdx.x, j = threadIdx.x;
    const float* dist = (const float*)(ws + OFF_DIST);

    sd[j] = dist[i * kB + j];
    st[j] = fabsf(targets[i] - targets[j]);
    __syncthreads();

    const float dij = sd[j];
    const float tij = st[j];
    float s = 0.0f;
    unsigned int c = 0;
    if (j != i) {                      // k==i, k==j auto-excluded by strict '<'
        for (int k = 0; k < kB; ++k) {
            const float tik = st[k];
            if (tij < tik) {
                ++c;
                const float v = dij + kMargin - sd[k];
                s += (v > 0.0f) ? v : 0.0f;
            }
        }
    }
    rs[j] = (double)s; rc[j] = c;
    __syncthreads();
    for (int o = 128; o > 0; o >>= 1) {
        if (j < o) { rs[j] += rs[j + o]; rc[j] += rc[j + o]; }
        __syncthreads();
    }
    if (j == 0) {
        atomicAdd((double*)(ws + OFF_SC), rs[0]);
        atomicAdd((unsigned int*)(ws + OFF_SC + 12), rc[0]);
    }
}

// Regression L1 + final combine. Single block.
__global__ __launch_bounds__(256) void k_final(const float* __restrict__ age,
                                               const float* __restrict__ targets,
                                               char* ws, float* __restrict__ out) {
    __shared__ float r[256];
    const int t = threadIdx.x;
    float a = 0.0f;
    for (int e = t; e < kS * kB; e += 256)
        a += fabsf(age[e] - targets[e & (kB - 1)]);
    r[t] = a;
    __syncthreads();
    for (int o = 128; o > 0; o >>= 1) {
        if (t < o) r[t] += r[t + o];
        __syncthreads();
    }
    if (t == 0) {
        const float reg = r[0] / (float)(kS * kB);
        const double ts = *(double*)(ws + OFF_SC);
        const float vs  = *(float*)(ws + OFF_SC + 8);
        const unsigned int ct = *(unsigned int*)(ws + OFF_SC + 12);
        const float ord = (ct > 0u) ? (float)(ts / (double)ct) : 0.0f;
        const float vib = 0.5f * vs / (float)kB;
        out[0] = reg + kBeta * ord + kAlpha * vib;
        out[1] = reg;
        out[2] = ord;
        out[3] = vib;
    }
}

extern "C" void kernel_launch(void* const* d_in, const int* in_sizes, int n_in,
                              void* d_out, int out_size, void* d_ws, size_t ws_size,
                              hipStream_t stream) {
    const float* age     = (const float*)d_in[0];  // [S,B]
    const float* mu      = (const float*)d_in[1];  // [B,D]
    const float* log_var = (const float*)d_in[2];  // [B,D]
    const float* targets = (const float*)d_in[3];  // [B]
    float* out = (float*)d_out;
    char* ws = (char*)d_ws;
    (void)in_sizes; (void)n_in; (void)out_size; (void)ws_size;

    k_init<<<1, 1, 0, stream>>>(ws);
    k_prep<<<kB, 256, 0, stream>>>(mu, log_var, ws);
    k_gemm<<<32, 256, 0, stream>>>(mu, ws);          // 256 waves, 256 WMMA tiles
    k_dist<<<kB, kB, 0, stream>>>(ws);
    k_trip<<<kB, kB, 0, stream>>>(targets, ws);
    k_final<<<1, 256, 0, stream>>>(age, targets, ws, out);
}